// SequenceMergingSeq_2671469658112
// MI455X (gfx1250) — compile-verified
//
#include <hip/hip_runtime.h>

#define B_   32
#define S_   2048
#define D_   512
#define T_   64      // tokens per chunk
#define NCH_ 32      // S_/T_

typedef __attribute__((ext_vector_type(2))) float v2f;
typedef __attribute__((ext_vector_type(4))) float v4f;
typedef __attribute__((ext_vector_type(8))) float v8f;

// ---------------------------------------------------------------------------
// K1: fused per-token stats (decay dot, exp-sum) + chunk-local linear scan.
// Scan inside each 16-token sub-block is a dense 16x16x16 GEMM via four
// chained V_WMMA_F32_16X16X4_F32 per 16-channel group.
// Global traffic: b128, non-temporal (read-once streams).
// ---------------------------------------------------------------------------
__global__ __launch_bounds__(512) void k_scan(
    const float* __restrict__ C, const float* __restrict__ V,
    const float* __restrict__ W, const float* __restrict__ wd,
    const float* __restrict__ bd,
    float* __restrict__ localB, float* __restrict__ cumP,
    float* __restrict__ aLoc)
{
  __shared__ float ev[T_ * D_];                  // 128 KB: ev, then local b
  __shared__ float dArr[T_], sArr[T_], Qsub[T_], cP[T_], aL[T_];
  __shared__ float Lm[16 * 16];

  const int tid  = threadIdx.x;
  const int lane = tid & 31;
  const int wv   = tid >> 5;        // 16 waves
  const int ch   = blockIdx.x;
  const int b    = blockIdx.y;
  const int t0   = ch * T_;
  const float bd0 = bd[0];

  // lane owns channels c = 4*lane + 128*i + j  (contiguous 512B per wave-load)
  v4f wdr[4];
  #pragma unroll
  for (int i = 0; i < 4; ++i)
    wdr[i] = *(const v4f*)(wd + 4 * lane + 128 * i);

  // ---- Phase 1: one wave per token; wave32 shfl reductions, ev -> LDS ----
  for (int j = 0; j < 4; ++j) {
    const int t = wv + 16 * j;
    const size_t base = ((size_t)b * S_ + t0 + t) * D_;
    float wsum = 0.f, esum = 0.f;
    #pragma unroll
    for (int i = 0; i < 4; ++i) {
      const int c0 = 4 * lane + 128 * i;
      const v4f cv = __builtin_nontemporal_load((const v4f*)(C + base + c0));
      const v4f vv = __builtin_nontemporal_load((const v4f*)(V + base + c0));
      const v4f wv4 = __builtin_nontemporal_load((const v4f*)(W + base + c0));
      v4f e;
      #pragma unroll
      for (int q = 0; q < 4; ++q) e[q] = __expf(cv[q]);
      esum += e[0] + e[1] + e[2] + e[3];
      *(v4f*)(&ev[t * D_ + c0]) = e * vv;
      const v4f wp = wv4 * wdr[i];
      wsum += wp[0] + wp[1] + wp[2] + wp[3];
    }
    #pragma unroll
    for (int off = 16; off > 0; off >>= 1) {
      wsum += __shfl_xor(wsum, off);
      esum += __shfl_xor(esum, off);
    }
    if (lane == 0) {
      dArr[t] = 1.f / (1.f + __expf(-(wsum + bd0)));
      sArr[t] = esum;
    }
  }
  __syncthreads();

  // ---- tiny serial section: decay products + scalar 'a' scan ----
  if (tid == 0) {
    float q = 1.f, a = 0.f, qs = 1.f;
    for (int t = 0; t < T_; ++t) {
      if ((t & 15) == 0) qs = 1.f;     // restart per 16-token sub-block
      const float d = dArr[t];
      qs *= d; Qsub[t] = qs;
      q  *= d; cP[t]   = q;            // chunk-level cumprod (flush-to-0 OK)
      a = d * a + sArr[t]; aL[t] = a;
    }
  }
  __syncthreads();

  const int half = lane >> 4;          // 0: lanes 0-15, 1: lanes 16-31
  const int l15  = lane & 15;

  // ---- Phase 2: scan as WMMA GEMMs, sub-blocks sequential ----
  for (int sb = 0; sb < 4; ++sb) {
    const int m0 = sb * 16;
    if (tid < 256) {                   // L[m][k] = prod_{j=k+1..m} d_j
      const int m = tid >> 4, k = tid & 15;
      Lm[tid] = (k <= m) ? __fdividef(Qsub[m0 + m], Qsub[m0 + k]) : 0.f;
    }
    __syncthreads();
    #pragma unroll
    for (int g2 = 0; g2 < 2; ++g2) {   // each wave owns 2 channel groups
      const int n0   = (wv * 2 + g2) * 16;
      const int cidx = n0 + l15;

      // C accumulator = carry term Q[m] * b_in[n]
      v8f acc;
      if (sb == 0) {
        #pragma unroll
        for (int v = 0; v < 8; ++v) acc[v] = 0.f;
      } else {
        const float bc = ev[(m0 - 1) * D_ + cidx];
        #pragma unroll
        for (int v = 0; v < 8; ++v) acc[v] = Qsub[m0 + v + half * 8] * bc;
      }

      // load ALL fragments before any in-place overwrite
      v2f af[4], bf[4];
      #pragma unroll
      for (int i = 0; i < 4; ++i) {
        const int kk = i * 4;
        af[i].x = Lm[l15 * 16 + kk + 2 * half];        // A: 16x4, K split 0,1|2,3
        af[i].y = Lm[l15 * 16 + kk + 2 * half + 1];
        bf[i].x = ev[(m0 + kk + 2 * half    ) * D_ + cidx];  // B: 4x16 rows
        bf[i].y = ev[(m0 + kk + 2 * half + 1) * D_ + cidx];
      }
      #pragma unroll
      for (int i = 0; i < 4; ++i)
        acc = __builtin_amdgcn_wmma_f32_16x16x4_f32(
            false, af[i], false, bf[i], (short)0, acc, false, false);

      #pragma unroll
      for (int v = 0; v < 8; ++v)      // D: vgpr v -> row v (+8 for hi lanes)
        ev[(m0 + v + half * 8) * D_ + cidx] = acc[v];
    }
    __syncthreads();
  }

  // ---- stream chunk-local results + per-token scalars to global (b128) ----
  for (int j = 0; j < 4; ++j) {
    const int t = wv + 16 * j;
    const size_t base = ((size_t)b * S_ + t0 + t) * D_;
    #pragma unroll
    for (int i = 0; i < 4; ++i) {
      const int c0 = 4 * lane + 128 * i;
      *(v4f*)(localB + base + c0) = *(const v4f*)(&ev[t * D_ + c0]);
    }
  }
  if (tid < T_) {
    cumP[(size_t)b * S_ + t0 + tid] = cP[tid];
    aLoc[(size_t)b * S_ + t0 + tid] = aL[tid];
  }
}

// ---------------------------------------------------------------------------
// K2: compose the linear recurrence across chunks (carry-in per chunk).
// Each thread owns 4 contiguous channels (b128 traffic).
// ---------------------------------------------------------------------------
__global__ __launch_bounds__(256) void k_carry(
    const float* __restrict__ localB, const float* __restrict__ cumP,
    const float* __restrict__ aLoc,
    float* __restrict__ carryA, float* __restrict__ carryB)
{
  const int idx = blockIdx.x * 256 + threadIdx.x;  // 0 .. B*D/4-1
  const int b  = idx / (D_ / 4);
  const int c0 = (idx % (D_ / 4)) * 4;
  v4f cb; cb[0] = cb[1] = cb[2] = cb[3] = 0.f;
  float ca = 0.f;
  for (int ch = 0; ch < NCH_; ++ch) {
    *(v4f*)(carryB + ((size_t)b * NCH_ + ch) * D_ + c0) = cb;
    if (c0 == 0) carryA[b * NCH_ + ch] = ca;
    const int tEnd = ch * T_ + T_ - 1;
    const float A  = cumP[(size_t)b * S_ + tEnd];
    const float aE = aLoc[(size_t)b * S_ + tEnd];
    const v4f  bE  = *(const v4f*)(localB + ((size_t)b * S_ + tEnd) * D_ + c0);
    cb = A * cb + bE;
    ca = A * ca + aE;
  }
}

// ---------------------------------------------------------------------------
// K3: apply carries, divide, fused LayerNorm (wave-per-token, barrier-free),
// in place on d_out. b128 loads (RT: likely L2-resident) + NT final stores.
// ---------------------------------------------------------------------------
__global__ __launch_bounds__(512) void k_final(
    float* __restrict__ io, const float* __restrict__ cumP,
    const float* __restrict__ aLoc,
    const float* __restrict__ carryA, const float* __restrict__ carryB,
    const float* __restrict__ gamma, const float* __restrict__ beta)
{
  const int tid  = threadIdx.x;
  const int lane = tid & 31;
  const int wv   = tid >> 5;
  const int ch   = blockIdx.x;
  const int b    = blockIdx.y;
  const int t0   = ch * T_;

  v4f gm[4], bt[4], cbv[4];
  #pragma unroll
  for (int i = 0; i < 4; ++i) {
    const int c0 = 4 * lane + 128 * i;
    gm[i]  = *(const v4f*)(gamma + c0);
    bt[i]  = *(const v4f*)(beta + c0);
    cbv[i] = *(const v4f*)(carryB + ((size_t)b * NCH_ + ch) * D_ + c0);
  }
  const float ca = carryA[b * NCH_ + ch];

  for (int j = 0; j < 4; ++j) {
    const int ts = t0 + wv + 16 * j;
    const size_t base = ((size_t)b * S_ + ts) * D_;
    const float cp  = cumP[(size_t)b * S_ + ts];
    const float inv = 1.f / (aLoc[(size_t)b * S_ + ts] + cp * ca + 1e-8f);
    v4f r[4]; float sum = 0.f, sq = 0.f;
    #pragma unroll
    for (int i = 0; i < 4; ++i) {
      const int c0 = 4 * lane + 128 * i;
      v4f x = *(const v4f*)(io + base + c0);
      x = (x + cp * cbv[i]) * inv;
      r[i] = x;
      sum += x[0] + x[1] + x[2] + x[3];
      const v4f x2 = x * x;
      sq  += x2[0] + x2[1] + x2[2] + x2[3];
    }
    #pragma unroll
    for (int off = 16; off > 0; off >>= 1) {
      sum += __shfl_xor(sum, off);
      sq  += __shfl_xor(sq,  off);
    }
    const float mu  = sum * (1.f / D_);
    const float var = sq * (1.f / D_) - mu * mu;
    const float sc  = rsqrtf(var + 1e-5f);
    #pragma unroll
    for (int i = 0; i < 4; ++i) {
      const int c0 = 4 * lane + 128 * i;
      const v4f o = (r[i] - mu) * sc * gm[i] + bt[i];
      __builtin_nontemporal_store(o, (v4f*)(io + base + c0));
    }
  }
}

extern "C" void kernel_launch(void* const* d_in, const int* in_sizes, int n_in,
                              void* d_out, int out_size, void* d_ws, size_t ws_size,
                              hipStream_t stream) {
  const float* C     = (const float*)d_in[0];
  const float* V     = (const float*)d_in[1];
  const float* W     = (const float*)d_in[2];
  const float* wd    = (const float*)d_in[3];
  const float* bd    = (const float*)d_in[4];
  const float* gamma = (const float*)d_in[5];
  const float* beta  = (const float*)d_in[6];
  float* out = (float*)d_out;

  // workspace layout (~2.6 MB): cumP[B*S], aLoc[B*S], carryA[B*NCH], carryB[B*NCH*D]
  float* cumP   = (float*)d_ws;
  float* aLoc   = cumP   + (size_t)B_ * S_;
  float* carryA = aLoc   + (size_t)B_ * S_;
  float* carryB = carryA + (size_t)B_ * NCH_;

  k_scan <<<dim3(NCH_, B_), 512, 0, stream>>>(C, V, W, wd, bd, out, cumP, aLoc);
  k_carry<<<(B_ * D_ / 4) / 256, 256, 0, stream>>>(out, cumP, aLoc, carryA, carryB);
  k_final<<<dim3(NCH_, B_), 512, 0, stream>>>(out, cumP, aLoc, carryA, carryB, gamma, beta);
}